// Cell_15642270892329
// MI455X (gfx1250) — compile-verified
//
#include <hip/hip_runtime.h>
#include <math.h>

// ---------------------------------------------------------------------------
// Problem constants (match the reference)
// ---------------------------------------------------------------------------
#define N_NODES 50000
#define NNZ_    800000
#define D_PREV  256
#define D_HID   128
#define LN_EPS  1e-5f

typedef __attribute__((ext_vector_type(2))) float v2f;
typedef __attribute__((ext_vector_type(8))) float v8f;

// ---------------------------------------------------------------------------
// Kernel 1: H = X @ W + b   via V_WMMA_F32_16X16X4_F32 (fp32 matrix core)
//   grid.x = N_NODES/16 row tiles, 256 threads = 8 waves, one 16-col tile/wave
//   A-frag (16x4 f32): lane m=row, VGPR0/1 hold K = k0+(half?2:0), +1
//   B-frag (4x16 f32): lane m=col, same K split across lane halves
//   C/D  (16x16 f32): VGPR r -> row r (lanes 0-15) / row r+8 (lanes 16-31)
// ---------------------------------------------------------------------------
__global__ __launch_bounds__(256) void gemm_wmma_f32(
    const float* __restrict__ X, const float* __restrict__ W,
    const float* __restrict__ bias, float* __restrict__ H)
{
  const int lane    = threadIdx.x & 31;
  const int wave    = threadIdx.x >> 5;     // 0..7 -> 16-column tile
  const int rowBase = blockIdx.x << 4;      // 16 rows per block
  const int half    = lane >> 4;            // lane half selects K pair
  const int m       = lane & 15;
  const int lrow    = rowBase + m;
  const int lcol    = (wave << 4) + m;

  const float* __restrict__ xrow = X + (size_t)lrow * D_PREV;

  v8f acc = {};
  #pragma unroll 8
  for (int k0 = 0; k0 < D_PREV; k0 += 4) {
    const int ka = k0 + (half << 1);
    v2f a = *(const v2f*)(xrow + ka);                 // K=ka, ka+1 for this lane
    v2f bm;
    bm.x = W[(size_t)ka * D_HID + lcol];              // B row K=ka
    bm.y = W[(size_t)(ka + 1) * D_HID + lcol];        // B row K=ka+1
    acc = __builtin_amdgcn_wmma_f32_16x16x4_f32(
        /*neg_a=*/false, a, /*neg_b=*/false, bm,
        /*c_mod=*/(short)0, acc, /*reuse_a=*/false, /*reuse_b=*/false);
  }

  const float bv = bias[lcol];
  #pragma unroll
  for (int r = 0; r < 8; ++r) {
    const int row = rowBase + r + (half << 3);
    H[(size_t)row * D_HID + lcol] = acc[r] + bv;
  }
}

// ---------------------------------------------------------------------------
// Kernel 2: zero-fill the accumulation states (float4 stores)
// ---------------------------------------------------------------------------
__global__ __launch_bounds__(256) void zero_f4(float4* __restrict__ p, long long n4)
{
  const long long i = (long long)blockIdx.x * blockDim.x + threadIdx.x;
  if (i < n4) p[i] = make_float4(0.f, 0.f, 0.f, 0.f);
}

// ---------------------------------------------------------------------------
// Kernel 3: edge-parallel SpMM scatter:  Y[rows[e]] += vals[e] * X[cols[e]]
//   one wave per edge; lane handles a float4 column chunk (32*4 = 128 cols)
//   gather is a fully coalesced 512B row read; scatter uses native f32
//   atomics resolved in L2 (output fits in 192MB L2).
//   Adjacency selector read on-device from idx_arr[idx_pos] (scalar broadcast).
// ---------------------------------------------------------------------------
__global__ __launch_bounds__(256) void spmm_scatter(
    const int*   __restrict__ adj_rows, const int* __restrict__ adj_cols,
    const float* __restrict__ adj_vals, const int* __restrict__ idx_arr,
    int idx_pos, const float* __restrict__ Xs, float* __restrict__ Y)
{
  const long long t = (long long)blockIdx.x * blockDim.x + threadIdx.x;
  const long long e = t >> 5;
  if (e >= NNZ_) return;
  const int lane = threadIdx.x & 31;

  const int    si   = idx_arr[idx_pos];               // uniform -> SMEM load
  const size_t base = (size_t)si * NNZ_ + (size_t)e;
  const int    r    = adj_rows[base];
  const int    c    = adj_cols[base];
  const float  v    = adj_vals[base];

  const float4 xv = *(const float4*)(Xs + (size_t)c * D_HID + (lane << 2));
  float* yp = Y + (size_t)r * D_HID + (lane << 2);
  __hip_atomic_fetch_add(yp + 0, v * xv.x, __ATOMIC_RELAXED, __HIP_MEMORY_SCOPE_AGENT);
  __hip_atomic_fetch_add(yp + 1, v * xv.y, __ATOMIC_RELAXED, __HIP_MEMORY_SCOPE_AGENT);
  __hip_atomic_fetch_add(yp + 2, v * xv.z, __ATOMIC_RELAXED, __HIP_MEMORY_SCOPE_AGENT);
  __hip_atomic_fetch_add(yp + 3, v * xv.w, __ATOMIC_RELAXED, __HIP_MEMORY_SCOPE_AGENT);
}

// ---------------------------------------------------------------------------
// Kernel 4: fused LayerNorm (eps=1e-5) + exact-erf GELU, one wave32 per row
// ---------------------------------------------------------------------------
__global__ __launch_bounds__(256) void ln_gelu(
    const float* __restrict__ Y, const float* __restrict__ gamma,
    const float* __restrict__ beta, float* __restrict__ Out)
{
  const int lane = threadIdx.x & 31;
  const int wave = threadIdx.x >> 5;
  const int row  = (blockIdx.x << 3) + wave;

  const float4 v = *(const float4*)(Y + (size_t)row * D_HID + (lane << 2));
  float s = v.x + v.y + v.z + v.w;
  float q = v.x * v.x + v.y * v.y + v.z * v.z + v.w * v.w;
  #pragma unroll
  for (int off = 16; off > 0; off >>= 1) {     // wave32 tree reduce
    s += __shfl_xor(s, off, 32);
    q += __shfl_xor(q, off, 32);
  }
  const float mu   = s * (1.0f / D_HID);
  const float var  = q * (1.0f / D_HID) - mu * mu;
  const float rstd = rsqrtf(var + LN_EPS);

  const float4 g  = *(const float4*)(gamma + (lane << 2));
  const float4 be = *(const float4*)(beta  + (lane << 2));
  const float vv[4] = {v.x, v.y, v.z, v.w};
  const float gg[4] = {g.x, g.y, g.z, g.w};
  const float bb[4] = {be.x, be.y, be.z, be.w};
  float o[4];
  #pragma unroll
  for (int j = 0; j < 4; ++j) {
    const float t = (vv[j] - mu) * rstd * gg[j] + bb[j];
    o[j] = 0.5f * t * (1.0f + erff(t * 0.70710678118654752f));
  }
  *(float4*)(Out + (size_t)row * D_HID + (lane << 2)) =
      make_float4(o[0], o[1], o[2], o[3]);
}

// ---------------------------------------------------------------------------
// Launch: s0 = x@W+b ; s1 = A[seq0] s0 ; s2 = A[seq1] s1 + A[res0] s0 ;
//         s3 = A[seq2] s2 + A[res1] s0 + A[res2] s1 ; out = gelu(LN(s3))
// ---------------------------------------------------------------------------
extern "C" void kernel_launch(void* const* d_in, const int* in_sizes, int n_in,
                              void* d_out, int out_size, void* d_ws, size_t ws_size,
                              hipStream_t stream) {
  const float* x         = (const float*)d_in[0];
  const int*   adj_rows  = (const int*)  d_in[1];
  const int*   adj_cols  = (const int*)  d_in[2];
  const float* adj_vals  = (const float*)d_in[3];
  const int*   idxes_seq = (const int*)  d_in[4];
  const int*   idxes_res = (const int*)  d_in[5];
  const float* W         = (const float*)d_in[6];
  const float* b         = (const float*)d_in[7];
  const float* gamma     = (const float*)d_in[8];
  const float* beta      = (const float*)d_in[9];
  float*       out       = (float*)d_out;

  const size_t stateElems = (size_t)N_NODES * D_HID;   // 6.4M floats
  float* s0 = (float*)d_ws;
  float* s1 = s0 + stateElems;
  float* s2 = s1 + stateElems;
  float* s3 = s2 + stateElems;

  // Zero accumulation targets s1..s3 (s0 fully overwritten by GEMM).
  const long long zero4 = (long long)(3 * stateElems) / 4;       // 4.8M float4
  zero_f4<<<(unsigned)((zero4 + 255) / 256), 256, 0, stream>>>((float4*)s1, zero4);

  // GEMM: 50000/16 = 3125 row tiles.
  gemm_wmma_f32<<<N_NODES / 16, 256, 0, stream>>>(x, W, b, s0);

  // SpMM chain: NNZ waves -> NNZ*32 threads = 100000 blocks of 256.
  const unsigned spmmBlocks = (unsigned)(((long long)NNZ_ * 32 + 255) / 256);
  // step 0
  spmm_scatter<<<spmmBlocks, 256, 0, stream>>>(adj_rows, adj_cols, adj_vals,
                                               idxes_seq, 0, s0, s1);
  // step 1
  spmm_scatter<<<spmmBlocks, 256, 0, stream>>>(adj_rows, adj_cols, adj_vals,
                                               idxes_seq, 1, s1, s2);
  spmm_scatter<<<spmmBlocks, 256, 0, stream>>>(adj_rows, adj_cols, adj_vals,
                                               idxes_res, 0, s0, s2);
  // step 2
  spmm_scatter<<<spmmBlocks, 256, 0, stream>>>(adj_rows, adj_cols, adj_vals,
                                               idxes_seq, 2, s2, s3);
  spmm_scatter<<<spmmBlocks, 256, 0, stream>>>(adj_rows, adj_cols, adj_vals,
                                               idxes_res, 1, s0, s3);
  spmm_scatter<<<spmmBlocks, 256, 0, stream>>>(adj_rows, adj_cols, adj_vals,
                                               idxes_res, 2, s1, s3);

  // LayerNorm + exact GELU: 8 rows per block.
  ln_gelu<<<N_NODES / 8, 256, 0, stream>>>(s3, gamma, beta, out);
}